// _GaussianFieldComponent2D_58652073394177
// MI455X (gfx1250) — compile-verified
//
#include <hip/hip_runtime.h>

typedef __attribute__((ext_vector_type(2))) float v2f;
typedef __attribute__((ext_vector_type(8))) float v8f;

#define HH 2048
#define WW 2048
#define NGAUSS 65536
#define RADIUS 16
#define PATCH 33              // 2R+1
#define HP (HH + 2 * RADIUS)  // 2080
#define WP (WW + 2 * RADIUS)  // 2080

__device__ __forceinline__ float softplus_f(float x) {
    // numerically stable softplus: max(x,0) + log1p(exp(-|x|))
    return fmaxf(x, 0.0f) + log1pf(__expf(-fabsf(x)));
}

// ---------------------------------------------------------------------------
// Zero the padded accumulator planes (fr, fi) in workspace.
// ---------------------------------------------------------------------------
__global__ __launch_bounds__(256) void gf_zero_kernel(float4* __restrict__ p, int n4) {
    const float4 z = make_float4(0.f, 0.f, 0.f, 0.f);
    for (int i = blockIdx.x * blockDim.x + threadIdx.x; i < n4; i += gridDim.x * blockDim.x)
        p[i] = z;
}

// ---------------------------------------------------------------------------
// Splat: one wave32 per Gaussian. Exponent of the Gaussian over a 16x16 pixel
// tile is a rank-3 product  [1, dy, dy^2] x [a*dx^2, b*dx, g]^T  evaluated by
// V_WMMA_F32_16X16X4_F32. 3x3 tiles cover the 33x33 patch; out-of-patch
// elements are masked only at the atomic (EXEC all-1s at the WMMA).
// log2(e) is folded into the quadratic coefficients; g = raw v_exp_f32
// (tail underflow flushes to 0, which is the desired Gaussian behavior).
// ---------------------------------------------------------------------------
__global__ __launch_bounds__(256) void gf_splat_kernel(
    const float* __restrict__ means, const float* __restrict__ chol,
    const float* __restrict__ weights,
    float* __restrict__ fr, float* __restrict__ fi)
{
    const int lane = threadIdx.x & 31;
    const int wid  = threadIdx.x >> 5;
    const int gi   = blockIdx.x * 8 + wid;   // wave-uniform -> scalar loads

    const float mx = means[2 * gi + 0];
    const float my = means[2 * gi + 1];
    const float c0 = chol[3 * gi + 0];
    const float c1 = chol[3 * gi + 1];
    const float c2 = chol[3 * gi + 2];
    const float wr = weights[2 * gi + 0];
    const float wi = weights[2 * gi + 1];

    const float l11 = 0.5f + softplus_f(c0);
    const float l22 = 0.5f + softplus_f(c2);
    const float ia  = 1.0f / l11;            // a
    const float ic  = 1.0f / l22;            // c
    const float ib  = c1 * ia * ic;          // b = l21/(l11*l22)

    // e(y,x) = -0.5*[(a^2+b^2) dx^2 - 2bc dx dy + c^2 dy^2]; fold log2(e):
    const float L2E   = 1.4426950408889634f;
    const float alpha = -0.5f * L2E * (ia * ia + ib * ib);
    const float beta  =         L2E * (ib * ic);
    const float gamma = -0.5f * L2E * (ic * ic);

    const int itlx = (int)floorf(mx) - RADIUS;
    const int itly = (int)floorf(my) - RADIUS;

    const int  mo = lane & 15;
    const bool hi = lane >= 16;

    #pragma unroll
    for (int tr = 0; tr < 3; ++tr) {
        const float dy = (float)(itly + tr * 16 + mo) - my;
        // A (16x4 f32): lanes 0-15 hold K0,K1 = {1, dy}; lanes 16-31 K2,K3 = {dy^2, 0}
        v2f afrag;
        afrag.x = hi ? dy * dy : 1.0f;
        afrag.y = hi ? 0.0f    : dy;

        #pragma unroll
        for (int tc = 0; tc < 3; ++tc) {
            const float dx = (float)(itlx + tc * 16 + mo) - mx;
            // B (4x16 f32): lanes 0-15 hold K0,K1 = {alpha*dx^2, beta*dx}; 16-31 K2,K3 = {gamma, 0}
            v2f bfrag;
            bfrag.x = hi ? gamma : alpha * dx * dx;
            bfrag.y = hi ? 0.0f  : beta * dx;

            v8f acc = {};
            acc = __builtin_amdgcn_wmma_f32_16x16x4_f32(
                /*neg_a=*/false, afrag, /*neg_b=*/false, bfrag,
                /*c_mod=*/(short)0, acc, /*reuse_a=*/false, /*reuse_b=*/false);

            // px < 33: trivially true for tc<2; for tc==2 only column mo==0 (px==32).
            const bool col_ok = (tc < 2) || (mo == 0);
            const int  px     = tc * 16 + mo;          // D: N = lane&15
            const int  mbase  = hi ? 8 : 0;

            #pragma unroll
            for (int v = 0; v < 8; ++v) {
                // py < 33: trivially true for tr<2; for tr==2 only row mbase+v==0.
                const bool row_ok = (tr < 2) || (!hi && v == 0);
                if (col_ok && row_ok) {
                    const int py = tr * 16 + mbase + v;  // D: M = v + (lane>=16 ? 8 : 0)
                    const float g = __builtin_amdgcn_exp2f(acc[v]);  // raw v_exp_f32
                    // padded coords always in [0, 2080)
                    const int idx = (itly + py + RADIUS) * WP + (itlx + px + RADIUS);
                    __hip_atomic_fetch_add(&fr[idx], wr * g,
                                           __ATOMIC_RELAXED, __HIP_MEMORY_SCOPE_AGENT);
                    __hip_atomic_fetch_add(&fi[idx], wi * g,
                                           __ATOMIC_RELAXED, __HIP_MEMORY_SCOPE_AGENT);
                }
            }
        }
    }
}

// ---------------------------------------------------------------------------
// Composite: out = init + s * field, interleaved complex64 (float2).
// ---------------------------------------------------------------------------
__global__ __launch_bounds__(256) void gf_composite_kernel(
    const float* __restrict__ init_re, const float* __restrict__ init_im,
    const float* __restrict__ rscale,
    const float* __restrict__ fr, const float* __restrict__ fi,
    float2* __restrict__ out)
{
    const float s = rscale[0];
    const int n = HH * WW;
    for (int i = blockIdx.x * blockDim.x + threadIdx.x; i < n; i += gridDim.x * blockDim.x) {
        const int y = i >> 11;          // W = 2048
        const int x = i & (WW - 1);
        const int pidx = (y + RADIUS) * WP + (x + RADIUS);
        float2 o;
        o.x = init_re[i] + s * fr[pidx];
        o.y = init_im[i] + s * fi[pidx];
        out[i] = o;
    }
}

extern "C" void kernel_launch(void* const* d_in, const int* in_sizes, int n_in,
                              void* d_out, int out_size, void* d_ws, size_t ws_size,
                              hipStream_t stream) {
    const float* means   = (const float*)d_in[0];
    const float* chol    = (const float*)d_in[1];
    const float* weights = (const float*)d_in[2];
    const float* init_re = (const float*)d_in[3];
    const float* init_im = (const float*)d_in[4];
    const float* rscale  = (const float*)d_in[5];

    float* fr = (float*)d_ws;                 // HP*WP floats
    float* fi = fr + (size_t)HP * WP;         // HP*WP floats (total 34.6 MB, fits L2)

    // 1) zero accumulators
    const int n4 = (2 * HP * WP) / 4;
    gf_zero_kernel<<<4096, 256, 0, stream>>>((float4*)d_ws, n4);

    // 2) WMMA splat: 8 Gaussians (8 waves) per 256-thread block
    gf_splat_kernel<<<NGAUSS / 8, 256, 0, stream>>>(means, chol, weights, fr, fi);

    // 3) composite to interleaved complex64
    gf_composite_kernel<<<4096, 256, 0, stream>>>(init_re, init_im, rscale, fr, fi,
                                                  (float2*)d_out);
}